// GATLayer_71811853189759
// MI455X (gfx1250) — compile-verified
//
#include <hip/hip_runtime.h>
#include <hip/hip_bf16.h>

typedef __attribute__((ext_vector_type(2))) float v2f;
typedef __attribute__((ext_vector_type(8))) float v8f;

#define IN_DIM    128
#define OUT_DIM   128
#define NUM_HEADS 4
#define HEAD_DIM  32
#define NEG_SLOPE 0.2f
#define GAT_EPS   1e-16f

// ---------------------------------------------------------------------------
// Kernel 1: H = X * W using V_WMMA_F32_16X16X4_F32 (fp32 WMMA, exact math).
// Block = 256 threads = 8 waves. Each block computes a 16-row slab of H
// (16 x 128); wave w computes the 16x16 tile at columns [16w, 16w+16).
// A (16x4 fp32): lane L<16 holds A[M=L, K+0..1]; lane L>=16 holds A[M=L-16, K+2..3].
// B (4x16 fp32): lane L<16 holds B[K+0][n],B[K+1][n]; lane L>=16 holds B[K+2][n],B[K+3][n].
// C/D (16x16 f32, 8 VGPRs): VGPR r: lanes0-15 -> M=r, lanes16-31 -> M=r+8; N=lane&15.
// ---------------------------------------------------------------------------
__global__ __launch_bounds__(256) void gat_gemm_wmma(const float* __restrict__ X,
                                                     const float* __restrict__ W,
                                                     float* __restrict__ H,
                                                     int M) {
  const int wave = threadIdx.x >> 5;
  const int lane = threadIdx.x & 31;
  const int m0 = blockIdx.x * 16;
  const int n0 = wave * 16;
  const int rc   = lane & 15;          // row (for A) / col (for B,C)
  const int koff = (lane >> 4) * 2;    // 0 for lanes 0-15, 2 for lanes 16-31

  int arow = m0 + rc;
  if (arow >= M) arow = M - 1;         // clamp loads; EXEC must stay all-ones for WMMA
  const float* __restrict__ xrow = X + (size_t)arow * IN_DIM + koff;

  v8f c = {};
#pragma unroll
  for (int k = 0; k < IN_DIM; k += 4) {
    v2f a = *(const v2f*)(xrow + k);                       // A[M=rc, k+koff .. k+koff+1]
    v2f b;
    b.x = W[(size_t)(k + koff) * OUT_DIM + n0 + rc];       // B[k+koff][n]
    b.y = W[(size_t)(k + koff + 1) * OUT_DIM + n0 + rc];   // B[k+koff+1][n]
    c = __builtin_amdgcn_wmma_f32_16x16x4_f32(false, a, false, b, (short)0, c, false, false);
  }

  const int rbase = (lane >> 4) * 8;
#pragma unroll
  for (int r = 0; r < 8; ++r) {
    const int row = m0 + rbase + r;
    if (row < M) H[(size_t)row * OUT_DIM + n0 + rc] = c[r];
  }
}

// ---------------------------------------------------------------------------
// Kernel 2: per-node attention scalars.
//   alpha_src[n,h] = sum_d H[n, h*32+d] * a[h, d]
//   alpha_dst[n,h] = sum_d H[n, h*32+d] * a[h, 32+d]
// One wave per node; lane L covers dims [4L, 4L+4), head = L>>3; 8-lane xor-tree.
// ---------------------------------------------------------------------------
__global__ __launch_bounds__(256) void gat_alpha(const float* __restrict__ H,
                                                 const float* __restrict__ a,
                                                 float* __restrict__ alpha_src,
                                                 float* __restrict__ alpha_dst,
                                                 int N) {
  const int lane = threadIdx.x & 31;
  const int node = blockIdx.x * 8 + (threadIdx.x >> 5);
  if (node >= N) return;

  const int d0    = lane * 4;
  const int head  = lane >> 3;
  const int local = d0 & (HEAD_DIM - 1);

  const float4 hv = *(const float4*)(H + (size_t)node * OUT_DIM + d0);
  const float4 as = *(const float4*)(a + head * (2 * HEAD_DIM) + local);
  const float4 ad = *(const float4*)(a + head * (2 * HEAD_DIM) + HEAD_DIM + local);

  float ps = hv.x * as.x + hv.y * as.y + hv.z * as.z + hv.w * as.w;
  float pd = hv.x * ad.x + hv.y * ad.y + hv.z * ad.z + hv.w * ad.w;

#pragma unroll
  for (int off = 1; off <= 4; off <<= 1) {
    ps += __shfl_xor(ps, off, 32);
    pd += __shfl_xor(pd, off, 32);
  }
  if ((lane & 7) == 0) {
    alpha_src[(size_t)node * NUM_HEADS + head] = ps;
    alpha_dst[(size_t)node * NUM_HEADS + head] = pd;
  }
}

// ---------------------------------------------------------------------------
// Kernel 3: per-edge scores -> leaky_relu -> exp; scatter-add denominator.
// One thread per edge, all 4 heads as float4.
// ---------------------------------------------------------------------------
__global__ __launch_bounds__(256) void gat_edge_scores(const int* __restrict__ ei,
                                                       const float* __restrict__ alpha_src,
                                                       const float* __restrict__ alpha_dst,
                                                       float* __restrict__ exp_s,
                                                       float* __restrict__ sum_exp,
                                                       int E) {
  const int e = blockIdx.x * blockDim.x + threadIdx.x;
  if (e >= E) return;
  const int src = ei[e];
  const int dst = ei[E + e];

  const float4 s1 = *(const float4*)(alpha_src + (size_t)src * NUM_HEADS);
  const float4 s2 = *(const float4*)(alpha_dst + (size_t)dst * NUM_HEADS);

  float s[4] = {s1.x + s2.x, s1.y + s2.y, s1.z + s2.z, s1.w + s2.w};
  float4 ex;
#pragma unroll
  for (int h = 0; h < 4; ++h) {
    float v = s[h];
    v = (v >= 0.0f) ? v : NEG_SLOPE * v;
    ((float*)&ex)[h] = expf(v);
  }
  *(float4*)(exp_s + (size_t)e * NUM_HEADS) = ex;

  float* sp = sum_exp + (size_t)dst * NUM_HEADS;
  unsafeAtomicAdd(sp + 0, ex.x);
  unsafeAtomicAdd(sp + 1, ex.y);
  unsafeAtomicAdd(sp + 2, ex.z);
  unsafeAtomicAdd(sp + 3, ex.w);
}

// ---------------------------------------------------------------------------
// Kernel 4: out[dst] += h[src] * attn. One wave per edge; lane L handles the
// float4 at dim 4L (head = L>>3). h and out both live in the 192MB L2.
// ---------------------------------------------------------------------------
__global__ __launch_bounds__(256) void gat_aggregate(const int* __restrict__ ei,
                                                     const float* __restrict__ H,
                                                     const float* __restrict__ exp_s,
                                                     const float* __restrict__ sum_exp,
                                                     float* __restrict__ out,
                                                     int E) {
  const int lane = threadIdx.x & 31;
  const int e = blockIdx.x * 8 + (threadIdx.x >> 5);
  if (e >= E) return;

  const int src = ei[e];
  const int dst = ei[E + e];
  const int head = lane >> 3;

  const float attn = exp_s[(size_t)e * NUM_HEADS + head] /
                     (sum_exp[(size_t)dst * NUM_HEADS + head] + GAT_EPS);

  const int d0 = lane * 4;
  const float4 hv = *(const float4*)(H + (size_t)src * OUT_DIM + d0);
  float* op = out + (size_t)dst * OUT_DIM + d0;
  unsafeAtomicAdd(op + 0, hv.x * attn);
  unsafeAtomicAdd(op + 1, hv.y * attn);
  unsafeAtomicAdd(op + 2, hv.z * attn);
  unsafeAtomicAdd(op + 3, hv.w * attn);
}

// ---------------------------------------------------------------------------
extern "C" void kernel_launch(void* const* d_in, const int* in_sizes, int n_in,
                              void* d_out, int out_size, void* d_ws, size_t ws_size,
                              hipStream_t stream) {
  const float* X  = (const float*)d_in[0];  // [N, 128]
  const int*   ei = (const int*)d_in[1];    // [2, E]
  const float* W  = (const float*)d_in[2];  // [128, 128]
  const float* a  = (const float*)d_in[3];  // [4, 64]

  const int N = in_sizes[0] / IN_DIM;
  const int E = in_sizes[1] / 2;

  // Workspace layout (N=100000, E=1.6M -> ~82 MB total)
  float* H         = (float*)d_ws;                         // N*128
  float* alpha_src = H + (size_t)N * OUT_DIM;              // N*4
  float* alpha_dst = alpha_src + (size_t)N * NUM_HEADS;    // N*4
  float* exp_s     = alpha_dst + (size_t)N * NUM_HEADS;    // E*4
  float* sum_exp   = exp_s + (size_t)E * NUM_HEADS;        // N*4

  hipMemsetAsync(sum_exp, 0, (size_t)N * NUM_HEADS * sizeof(float), stream);
  hipMemsetAsync(d_out, 0, (size_t)out_size * sizeof(float), stream);

  // 1. H = X * W (WMMA fp32)
  gat_gemm_wmma<<<(N + 15) / 16, 256, 0, stream>>>(X, W, H, N);
  // 2. per-node alpha scalars
  gat_alpha<<<(N + 7) / 8, 256, 0, stream>>>(H, a, alpha_src, alpha_dst, N);
  // 3. edge scores + softmax denominator
  gat_edge_scores<<<(E + 255) / 256, 256, 0, stream>>>(ei, alpha_src, alpha_dst,
                                                       exp_s, sum_exp, E);
  // 4. weighted aggregation
  gat_aggregate<<<(E + 7) / 8, 256, 0, stream>>>(ei, H, exp_s, sum_exp,
                                                 (float*)d_out, E);
}